// ConvolutionalLayer_88742614270062
// MI455X (gfx1250) — compile-verified
//
#include <hip/hip_runtime.h>

// V_WMMA_F32_16X16X4_F32 operand types (wave32):
//   A: 16x4 f32 -> v2f (2 VGPRs/lane), B: 4x16 f32 -> v2f, C/D: 16x16 f32 -> v8f
typedef __attribute__((ext_vector_type(2))) float v2f;
typedef __attribute__((ext_vector_type(8))) float v8f;

#define NBATCH 32
#define C_IN   3
#define H_IN   224
#define W_IN   224
#define C_OUT  64
#define H_OUT  220   // 224 - 5 + 1
#define W_OUT  220
#define HT     14    // h-tiles; last tile overlaps (h0 = 204)
#define WC     10    // w columns per wave
#define NWC    22    // 220 / WC
#define PLANE  (W_OUT * H_OUT)   // 48400
#define KSLOTS 40    // kernel rows re-laid-out as 8 slots: [0 0 0 k0 k1 k2 k3 k4]

// out[n,o,w,h] = sum_{ki,kj} x[n,2,h+ki,w+kj] * ker[o,ki,kj]
// One wave32 computes 16(outch) x 16(h), looping over WC consecutive w.
// K-slot t = 4c + 2*half + j  ->  ki = t>>3, column offset = (t&7) - 3.
// Left-padding makes every junk load land in columns w-3..w-1 (inside the
// image for all n since channel 2 is preceded by channels 0/1), and the max
// index is exactly (223,223): all loads in-bounds for every tile, so there is
// no edge path at all. The last h-tile starts at 204 and overlap-writes rows
// 204..207 with identical values (benign).
__global__ __launch_bounds__(256) void ConvWmmaF32(const float* __restrict__ inp,
                                                   const float* __restrict__ ker,
                                                   float* __restrict__ out) {
  __shared__ float wpad[C_OUT * KSLOTS];   // 10 KB padded weights

  // Cooperative padded-weight fill (once per block).
  for (int idx = threadIdx.x; idx < C_OUT * KSLOTS; idx += 256) {
    const int o = idx / KSLOTS, s = idx % KSLOTS;
    const int ki = s >> 3, sl = s & 7;
    wpad[idx] = (sl >= 3) ? ker[o * 25 + ki * 5 + (sl - 3)] : 0.0f;
  }
  __syncthreads();

  const int lane = threadIdx.x & 31;
  const int wave = threadIdx.x >> 5;
  const int half = lane >> 4;   // 0: lanes 0-15 (K0/K1), 1: lanes 16-31 (K2/K3)
  const int lid  = lane & 15;   // M for A, N (=h) for B/D

  int t = blockIdx.x * 8 + wave;          // grid sized exactly
  const int wc = t % NWC; t /= NWC;
  const int ht = t % HT;  t /= HT;
  const int ot = t & 3;   t >>= 2;
  const int n  = t;
  int h0 = ht << 4;
  if (h0 > H_OUT - 16) h0 = H_OUT - 16;   // overlapped last tile: h0 = 204
  const int w0 = wc * WC;

  // A fragments from LDS: wrow[c*4 + j] == slot t = 4c + 2*half + j.
  v2f A[10];
  {
    const float* wrow = &wpad[(ot * 16 + lid) * KSLOTS + 2 * half];
#pragma unroll
    for (int c = 0; c < 10; ++c) {        // 10x ds_load_b64
      A[c].x = wrow[c * 4 + 0];
      A[c].y = wrow[c * 4 + 1];
    }
  }

  const int h = h0 + lid;
  // Per-lane B base: absorbs row, w0, the 2*half K-split, and the -3 left pad.
  const float* __restrict__ p =
      inp + (size_t)(n * C_IN + 2) * (H_IN * W_IN) + (h * W_IN + w0 + 2 * half - 3);
  float* __restrict__ ob =
      out + ((size_t)(n * C_OUT + ot * 16 + 8 * half) * W_OUT + w0) * (size_t)H_OUT + h;

#pragma unroll 1
  for (int wi = 0; wi < WC; ++wi) {
    v8f acc = {};
#pragma unroll
    for (int c = 0; c < 10; ++c) {
      v2f B;
      B.x = p[(c >> 1) * W_IN + (c & 1) * 4 + 0];   // immediate offsets only
      B.y = p[(c >> 1) * W_IN + (c & 1) * 4 + 1];
      acc = __builtin_amdgcn_wmma_f32_16x16x4_f32(
          false, A[c], false, B, (short)0, acc, false, false);
    }
#pragma unroll
    for (int v = 0; v < 8; ++v)           // stream: output 396MB >> L2 192MB
      __builtin_nontemporal_store(acc[v], ob + (size_t)v * PLANE);
    p += 1;            // next w
    ob += H_OUT;
  }
}

extern "C" void kernel_launch(void* const* d_in, const int* in_sizes, int n_in,
                              void* d_out, int out_size, void* d_ws, size_t ws_size,
                              hipStream_t stream) {
  const float* inp = (const float*)d_in[0];   // (32,3,224,224) f32
  const float* ker = (const float*)d_in[1];   // (64,5,5) f32
  float* out = (float*)d_out;                 // (32,64,220,220) f32 (h fastest)

  const int waves = NBATCH * 4 * HT * NWC;    // 39424, multiple of 8
  dim3 grid(waves / 8), block(256);
  ConvWmmaF32<<<grid, block, 0, stream>>>(inp, ker, out);
}